// HungarianCost_62045097558505
// MI455X (gfx1250) — compile-verified
//
#include <hip/hip_runtime.h>
#include <stdint.h>

// Problem constants (match reference)
#define BB 32
#define QQ 900
#define CC 92
#define CP 96        // padded class stride (float4-friendly, pad cols are 0)
#define TILE 64
#define BIGV 100000000.0f

typedef uint32_t u32x4 __attribute__((ext_vector_type(4)));
typedef uint32_t u32x8 __attribute__((ext_vector_type(8)));

// ---------------- CDNA5 async global->LDS helpers (inline asm, ASYNCcnt) ---------------

__device__ __forceinline__ uint32_t lds_addr32(const void* p) {
  // generic pointer to __shared__: low 32 bits are the LDS byte offset
  return (uint32_t)(uintptr_t)p;
}
__device__ __forceinline__ void async_ld_b32(uint32_t lds, const void* g) {
  asm volatile("global_load_async_to_lds_b32 %0, %1, off"
               :: "v"(lds), "v"((uint64_t)(uintptr_t)g) : "memory");
}
__device__ __forceinline__ void async_ld_b128(uint32_t lds, const void* g) {
  asm volatile("global_load_async_to_lds_b128 %0, %1, off"
               :: "v"(lds), "v"((uint64_t)(uintptr_t)g) : "memory");
}
__device__ __forceinline__ void async_wait_all() {
  asm volatile("s_wait_asynccnt 0" ::: "memory");
}
__device__ __forceinline__ float rcp_fast(float x) { return __builtin_amdgcn_rcpf(x); }

// ---------------- CDNA5 Tensor Data Mover: 2D tile (rows x 96 f32) -> LDS --------------
// D# per CDNA5 ISA:
//   group0: [1:0]=count=1, [63:32]=lds_addr, [120:64]=global byte addr, [127:126]=type=2
//   group1: [17:16]=data_size(2=4B), [79:48]=tensor_dim0, [111:80]=tensor_dim1,
//           [127:112]=tile_dim0, [143:128]=tile_dim1, [207:160]=tensor_dim0_stride
// OOB rows (beyond tensor_dim1) return zeros -> safe for the final partial tile.
__device__ __forceinline__ void tdm_load_tile96(uint32_t lds, const void* gbase,
                                                uint32_t remaining_rows) {
  const uint64_t ga = (uint64_t)(uintptr_t)gbase;
  u32x4 g0;
  g0[0] = 1u;                                                  // count=1 (valid user D#)
  g0[1] = lds;                                                 // lds_addr
  g0[2] = (uint32_t)ga;                                        // global_addr[31:0]
  g0[3] = (uint32_t)((ga >> 32) & 0x01FFFFFFu) | (2u << 30);   // addr[56:32] | type=2
  u32x8 g1;
  g1[0] = (2u << 16);                                          // data_size=4B, mask=0
  g1[1] = (96u << 16);                                         // tensor_dim0=96 (lo16)
  g1[2] = (remaining_rows & 0xFFFFu) << 16;                    // dim0 hi=0 | dim1 lo16
  g1[3] = ((remaining_rows >> 16) & 0xFFFFu) | (96u << 16);    // dim1 hi16 | tile_dim0=96
  g1[4] = (uint32_t)TILE;                                      // tile_dim1=64, tile_dim2=0
  g1[5] = 96u;                                                 // tensor_dim0_stride lo32
  g1[6] = 0u;                                                  // stride hi | dim1_stride lo
  g1[7] = 0u;
  asm volatile("tensor_load_to_lds %0, %1" :: "s"(g0), "s"(g1) : "memory");
}

// ---------------- Kernel 1: row softmax, one wave32 per row, padded to 96 -------------

__global__ __launch_bounds__(256) void softmax_rows(const float* __restrict__ logits,
                                                    float* __restrict__ probs) {
  const int lane = threadIdx.x & 31;
  const int row  = blockIdx.x * 8 + (threadIdx.x >> 5);
  if (row >= BB * QQ) return;
  const float* src = logits + (size_t)row * CC;
  const bool v2 = (lane + 64) < CC;            // lanes 28..31 cover pad cols 92..95
  float x0 = src[lane];
  float x1 = src[lane + 32];
  float x2 = v2 ? src[lane + 64] : -3.4e38f;
  float m = fmaxf(x0, fmaxf(x1, x2));
  #pragma unroll
  for (int off = 16; off > 0; off >>= 1) m = fmaxf(m, __shfl_xor(m, off, 32));
  float e0 = __expf(x0 - m);
  float e1 = __expf(x1 - m);
  float e2 = v2 ? __expf(x2 - m) : 0.0f;
  float s = e0 + e1 + e2;
  #pragma unroll
  for (int off = 16; off > 0; off >>= 1) s += __shfl_xor(s, off, 32);
  const float inv = rcp_fast(s);
  float* dst = probs + (size_t)row * CP;
  dst[lane]      = e0 * inv;
  dst[lane + 32] = e1 * inv;
  dst[lane + 64] = v2 ? (e2 * inv) : 0.0f;     // zero the 4 pad columns
}

// ---------------- Kernel 2: 64x64 cost tile, 4x4 micro-tile per thread ----------------

template <bool FUSED>
__global__ __launch_bounds__(256) void cost_kernel(
    const float* __restrict__ logits,      // FUSED path only
    const float* __restrict__ probs,       // !FUSED path (precomputed in d_ws)
    const float* __restrict__ pred_boxes,  // [B,Q,4] cxcywh
    const float* __restrict__ gt_boxes,    // [B,Q,4] cxcywh
    const float* __restrict__ area,        // [B,Q]
    const int*   __restrict__ labels,      // [B,Q]
    float* __restrict__ out) {             // [B,Q,Q]
  __shared__ float s_prob[TILE * CP];      // 24.6 KB softmax rows for i-tile
  __shared__ float s_pb[TILE * 4];
  __shared__ float s_gb[TILE * 4];
  __shared__ float s_area[TILE];
  __shared__ int   s_lab[TILE];

  const int tid = threadIdx.x;
  const int b  = blockIdx.z;
  const int i0 = blockIdx.y * TILE;
  const int j0 = blockIdx.x * TILE;

  // ---- stage prob tile ----
  if (FUSED) {
    // fallback: stage raw logits (92-wide rows, not TDM-paddable) via async b32
    #pragma unroll 1
    for (int k = 0; k < 23; ++k) {                         // 23*256 = 64*92 logits
      int c = tid + k * 256;
      int r = c / CC, col = c - r * CC;
      int gi = min(i0 + r, QQ - 1);
      async_ld_b32(lds_addr32(&s_prob[r * CP + col]),
                   logits + ((size_t)b * QQ + gi) * CC + col);
    }
  } else {
    // TDM: one DMA per workgroup moves the 64x96 f32 tile. EXEC is ignored by
    // TENSOR ops, so gate with a *uniform scalar* branch: only wave 0 issues it.
    if (__builtin_amdgcn_readfirstlane(tid) == 0) {
      const uint32_t row0 = (uint32_t)(b * QQ + i0);
      tdm_load_tile96(lds_addr32(s_prob), probs + (size_t)row0 * CP,
                      (uint32_t)(BB * QQ) - row0);
    }
  }

  // ---- async stage small arrays (each wave takes exactly one branch: full EXEC) ----
  if (tid < 64) {                                          // waves 0,1
    int gi = min(i0 + tid, QQ - 1);
    async_ld_b128(lds_addr32(&s_pb[tid * 4]), pred_boxes + ((size_t)b * QQ + gi) * 4);
  } else if (tid < 128) {                                  // waves 2,3
    int r = tid - 64, gj = min(j0 + r, QQ - 1);
    async_ld_b128(lds_addr32(&s_gb[r * 4]), gt_boxes + ((size_t)b * QQ + gj) * 4);
  } else if (tid < 192) {                                  // waves 4,5
    int r = tid - 128, gj = min(j0 + r, QQ - 1);
    async_ld_b32(lds_addr32(&s_area[r]), area + (size_t)b * QQ + gj);
  } else {                                                 // waves 6,7
    int r = tid - 192, gj = min(j0 + r, QQ - 1);
    async_ld_b32(lds_addr32(&s_lab[r]), labels + (size_t)b * QQ + gj);
  }
  async_wait_all();                        // drain this wave's ASYNCcnt
  __builtin_amdgcn_s_wait_tensorcnt(0);    // drain TDM (no-op for waves 1..7)
  __syncthreads();                         // cross-wave visibility of LDS

  if (FUSED) {           // fallback: softmax in LDS (only if d_ws was too small)
    if (tid < TILE) {
      float* rowp = &s_prob[tid * CP];
      float m = -3.4e38f;
      for (int c = 0; c < CC; ++c) m = fmaxf(m, rowp[c]);
      float s = 0.0f;
      for (int c = 0; c < CC; ++c) { float e = __expf(rowp[c] - m); rowp[c] = e; s += e; }
      float inv = rcp_fast(s);
      for (int c = 0; c < CC; ++c) rowp[c] *= inv;
      rowp[CC] = rowp[CC + 1] = rowp[CC + 2] = rowp[CC + 3] = 0.0f;
    }
    __syncthreads();
  }

  // ---- per-thread 4x4 micro-tile, all operands hoisted to registers ----
  const int tx = tid & 15, ty = tid >> 4;

  float pcx[4], pcy[4], pw[4], ph[4], px0[4], py0[4], px1[4], py1[4], pa[4];
  int pbase[4];
  #pragma unroll
  for (int a = 0; a < 4; ++a) {
    int il = ty * 4 + a;
    pbase[a] = il * CP;
    float cx = s_pb[il * 4 + 0], cy = s_pb[il * 4 + 1];
    float w  = s_pb[il * 4 + 2], h  = s_pb[il * 4 + 3];
    pcx[a] = cx; pcy[a] = cy; pw[a] = w; ph[a] = h;
    px0[a] = cx - 0.5f * w; py0[a] = cy - 0.5f * h;
    px1[a] = cx + 0.5f * w; py1[a] = cy + 0.5f * h;
    pa[a] = (px1[a] - px0[a]) * (py1[a] - py0[a]);
  }
  float gcx[4], gcy[4], gw[4], gh[4], gx0[4], gy0[4], gx1[4], gy1[4], ga[4];
  int lab[4]; bool msk[4];
  #pragma unroll
  for (int a = 0; a < 4; ++a) {
    int jl = tx * 4 + a;
    float cx = s_gb[jl * 4 + 0], cy = s_gb[jl * 4 + 1];
    float w  = s_gb[jl * 4 + 2], h  = s_gb[jl * 4 + 3];
    gcx[a] = cx; gcy[a] = cy; gw[a] = w; gh[a] = h;
    gx0[a] = cx - 0.5f * w; gy0[a] = cy - 0.5f * h;
    gx1[a] = cx + 0.5f * w; gy1[a] = cy + 0.5f * h;
    ga[a] = (gx1[a] - gx0[a]) * (gy1[a] - gy0[a]);
    lab[a] = s_lab[jl];
    msk[a] = s_area[jl] > 0.0f;
  }

  float acc[4][4];
  #pragma unroll
  for (int ii = 0; ii < 4; ++ii) {
    #pragma unroll
    for (int jj = 0; jj < 4; ++jj) {
      // L1 in cxcywh space (abs folds into VALU source modifiers)
      float l1 = fabsf(pcx[ii] - gcx[jj]) + fabsf(pcy[ii] - gcy[jj])
               + fabsf(pw[ii]  - gw[jj])  + fabsf(ph[ii]  - gh[jj]);
      // GIoU in xyxy space
      float iw  = fmaxf(fminf(px1[ii], gx1[jj]) - fmaxf(px0[ii], gx0[jj]), 0.0f);
      float ih  = fmaxf(fminf(py1[ii], gy1[jj]) - fmaxf(py0[ii], gy0[jj]), 0.0f);
      float inter = iw * ih;
      float uni   = pa[ii] + ga[jj] - inter;
      float ew  = fmaxf(fmaxf(px1[ii], gx1[jj]) - fminf(px0[ii], gx0[jj]), 0.0f);
      float eh  = fmaxf(fmaxf(py1[ii], gy1[jj]) - fminf(py0[ii], gy0[jj]), 0.0f);
      float enc = ew * eh;
      float ngiou = (enc - uni) * rcp_fast(enc) - inter * rcp_fast(uni); // == -giou
      float pr = s_prob[pbase[ii] + lab[jj]];            // ds_load_b32 gather
      float c  = fmaf(0.25f, l1, ngiou) - pr;
      acc[ii][jj] = msk[jj] ? c : BIGV;
    }
  }

  // ---- aligned float4 stores (QQ % 4 == 0, so the vector guard is exact) ----
  const int j = j0 + tx * 4;
  #pragma unroll
  for (int a = 0; a < 4; ++a) {
    int i = i0 + ty * 4 + a;
    if (i < QQ && j < QQ) {
      float4 v = make_float4(acc[a][0], acc[a][1], acc[a][2], acc[a][3]);
      *(float4*)(out + ((size_t)b * QQ + i) * QQ + j) = v;
    }
  }
}

// ---------------- host launcher -------------------------------------------------------

extern "C" void kernel_launch(void* const* d_in, const int* in_sizes, int n_in,
                              void* d_out, int out_size, void* d_ws, size_t ws_size,
                              hipStream_t stream) {
  (void)in_sizes; (void)n_in; (void)out_size;
  const float* logits = (const float*)d_in[0];   // [B,Q,C]
  const float* pboxes = (const float*)d_in[1];   // [B,Q,4]
  const float* gboxes = (const float*)d_in[2];   // [B,Q,4]
  const float* areas  = (const float*)d_in[3];   // [B,Q]
  const int*   labels = (const int*)d_in[4];     // [B,Q]
  float* out = (float*)d_out;

  dim3 grid((QQ + TILE - 1) / TILE, (QQ + TILE - 1) / TILE, BB);  // 15 x 15 x 32
  dim3 block(256);

  const size_t need = (size_t)BB * QQ * CP * sizeof(float);       // 11.06 MB
  if (ws_size >= need) {
    float* probs = (float*)d_ws;
    softmax_rows<<<(BB * QQ) / 8, 256, 0, stream>>>(logits, probs);
    cost_kernel<false><<<grid, block, 0, stream>>>(logits, probs, pboxes, gboxes,
                                                   areas, labels, out);
  } else {
    cost_kernel<true><<<grid, block, 0, stream>>>(logits, nullptr, pboxes, gboxes,
                                                  areas, labels, out);
  }
}